// MultiHeadAttention_19834158973377
// MI455X (gfx1250) — compile-verified
//
#include <hip/hip_runtime.h>
#include <hip/hip_bf16.h>
#include <math.h>

typedef __attribute__((ext_vector_type(2))) float v2f;
typedef __attribute__((ext_vector_type(8))) float v8f;

#define B_   2
#define L_   2048
#define HID_ 512
#define NH_  8
#define DH_  64
#define BL_  (B_ * L_)

// LDS row paddings (floats): keep 8B/16B alignment and spread the 64 banks.
#define APAD  36   // A tile rows (row-major, float4 staged)
#define BTPAD 34   // B tile stored TRANSPOSED: BsT[n][k]
#define KPAD  68   // K tile rows (row-major)
#define VTPAD 18   // V tile stored TRANSPOSED: VldsT[n][k]
#define PPAD  18   // P tile rows

__device__ __forceinline__ v8f wmma_f32(v2f a, v2f b, v8f c) {
  // v_wmma_f32_16x16x4_f32 : D = A(16x4) x B(4x16) + C(16x16)
  return __builtin_amdgcn_wmma_f32_16x16x4_f32(
      /*neg_a=*/false, a, /*neg_b=*/false, b,
      /*c_mod=*/(short)0, c, /*reuse_a=*/false, /*reuse_b=*/false);
}

// ---------------------------------------------------------------------------
// 128x64 output tile GEMM core: C(128x64) += A(128xK) * B(Kx64)
// Block = 128 threads = 4 waves. Wave w owns two 16-row strips:
//   strip 0: rows [w*16, w*16+16),  strip 1: rows [64+w*16, 64+w*16+16).
// acc[s][j] = 16x16 subtile (strip s, columns [j*16, j*16+16)).
// B is staged transposed so every WMMA B-fragment is one contiguous b64 read.
// ---------------------------------------------------------------------------
__device__ __forceinline__ void gemm_tile_128x64(
    const float* __restrict__ A, int lda,
    const float* __restrict__ Bm, int ldb,
    int K, float* As, float* BsT, v8f acc[2][4]) {
  const int tid  = threadIdx.x;
  const int wave = tid >> 5;
  const int lane = tid & 31;
  const int lh   = lane & 15;   // lane % 16
  const int lg   = lane >> 4;   // lane group 0/1

  for (int k0 = 0; k0 < K; k0 += 32) {
    // --- stage A tile (128 x 32), row-major ---
    {
      const int c4 = (tid & 7) * 4;
      const int r0 = tid >> 3;                  // 0..15
#pragma unroll
      for (int it = 0; it < 8; ++it) {
        const int r = r0 + it * 16;
        const float4 v = *(const float4*)(A + (size_t)r * lda + k0 + c4);
        *(float4*)(As + r * APAD + c4) = v;
      }
    }
    // --- stage B tile (32 x 64) TRANSPOSED: BsT[n*BTPAD + k] ---
    {
      const int c4 = (tid & 15) * 4;            // n base
      const int r0 = tid >> 4;                  // 0..7 (k)
#pragma unroll
      for (int it = 0; it < 4; ++it) {
        const int r = r0 + it * 8;
        const float4 v = *(const float4*)(Bm + (size_t)(k0 + r) * ldb + c4);
        BsT[(c4 + 0) * BTPAD + r] = v.x;
        BsT[(c4 + 1) * BTPAD + r] = v.y;
        BsT[(c4 + 2) * BTPAD + r] = v.z;
        BsT[(c4 + 3) * BTPAD + r] = v.w;
      }
    }
    __syncthreads();

#pragma unroll
    for (int kb = 0; kb < 8; ++kb) {
      const int kd = kb * 4 + 2 * lg;
      // A fragments: a[i] = A[m][kd+i] (contiguous b64)
      const v2f a0 = *(const v2f*)(As + (wave * 16 + lh) * APAD + kd);
      const v2f a1 = *(const v2f*)(As + (64 + wave * 16 + lh) * APAD + kd);
#pragma unroll
      for (int j = 0; j < 4; ++j) {
        // B fragment: b[i] = B[kd+i][n] = BsT[n][kd+i] (contiguous b64)
        const v2f b = *(const v2f*)(BsT + (j * 16 + lh) * BTPAD + kd);
        acc[0][j] = wmma_f32(a0, b, acc[0][j]);
        acc[1][j] = wmma_f32(a1, b, acc[1][j]);
      }
    }
    __syncthreads();
  }
}

// ---------------------------------------------------------------------------
// Kernel 1: Q/K/V projections -> head-split layout [B, NH, L, DH].
// Q pre-scaled by DH^-0.5.
// ---------------------------------------------------------------------------
__global__ __launch_bounds__(128) void proj_kernel(
    const float* __restrict__ X,
    const float* __restrict__ Wq, const float* __restrict__ bq,
    const float* __restrict__ Wk, const float* __restrict__ bk,
    const float* __restrict__ Wv, const float* __restrict__ bv,
    float* __restrict__ oq, float* __restrict__ ok, float* __restrict__ ov) {
  __shared__ __align__(16) float As[128 * APAD];
  __shared__ __align__(16) float BsT[64 * BTPAD];

  const int ntile = blockIdx.x;  // 0..7  (N = 512)
  const int mtile = blockIdx.y;  // 0..31 (M = 4096, tile 128)
  const int which = blockIdx.z;  // 0=Q 1=K 2=V

  const float* W  = (which == 0) ? Wq : (which == 1) ? Wk : Wv;
  const float* bi = (which == 0) ? bq : (which == 1) ? bk : bv;
  float*       O  = (which == 0) ? oq : (which == 1) ? ok : ov;
  const float scale = (which == 0) ? 0.125f : 1.0f;   // DH^-0.5 = 1/8

  v8f acc[2][4] = {};
  gemm_tile_128x64(X + (size_t)mtile * 128 * HID_, HID_,
                   W + ntile * 64, HID_, HID_, As, BsT, acc);

  const int tid = threadIdx.x, wave = tid >> 5, lane = tid & 31;
  const int lh = lane & 15, lg = lane >> 4;
#pragma unroll
  for (int s = 0; s < 2; ++s) {
#pragma unroll
    for (int j = 0; j < 4; ++j) {
      const int gc = ntile * 64 + j * 16 + lh;   // hidden col
      const float bv_ = bi[gc];
      const int h = gc >> 6, e = gc & 63;
#pragma unroll
      for (int v = 0; v < 8; ++v) {
        const int gr = mtile * 128 + s * 64 + wave * 16 + v + 8 * lg;
        const int bb = gr >> 11, l = gr & (L_ - 1);
        O[(((size_t)bb * NH_ + h) * L_ + l) * DH_ + e] =
            (acc[s][j][v] + bv_) * scale;
      }
    }
  }
}

// ---------------------------------------------------------------------------
// Kernel 2: typed query transform qt[t,b,h] = qh[b,h] @ typed_weight[t,h]
// Batched GEMM, M=2048 N=64 K=64.  batch = t*16 + b*8 + h.
// ---------------------------------------------------------------------------
__global__ __launch_bounds__(128) void qt_kernel(
    const float* __restrict__ qh, const float* __restrict__ tw,
    float* __restrict__ qt) {
  __shared__ __align__(16) float As[128 * APAD];
  __shared__ __align__(16) float BsT[64 * BTPAD];

  const int mtile = blockIdx.x;   // 0..15 (M = 2048, tile 128)
  const int batch = blockIdx.y;   // 0..47
  const int t  = batch >> 4;
  const int bh = batch & 15;
  const int h  = bh & 7;

  v8f acc[2][4] = {};
  const float* A  = qh + ((size_t)bh * L_ + mtile * 128) * DH_;
  const float* Bm = tw + ((size_t)t * NH_ + h) * DH_ * DH_;
  gemm_tile_128x64(A, DH_, Bm, DH_, DH_, As, BsT, acc);

  const int tid = threadIdx.x, wave = tid >> 5, lane = tid & 31;
  const int lh = lane & 15, lg = lane >> 4;
  float* O = qt + ((size_t)batch * L_ + mtile * 128) * DH_;
#pragma unroll
  for (int s = 0; s < 2; ++s)
#pragma unroll
    for (int j = 0; j < 4; ++j)
#pragma unroll
      for (int v = 0; v < 8; ++v)
        O[(s * 64 + wave * 16 + v + 8 * lg) * DH_ + j * 16 + lh] =
            acc[s][j][v];
}

// ---------------------------------------------------------------------------
// Kernel 3: streaming typed attention with online softmax.
// Block: 4 waves, 64 query rows (16 per wave). Keys streamed in tiles of 16.
// ---------------------------------------------------------------------------
__global__ __launch_bounds__(128) void attn_kernel(
    const float* __restrict__ qt, const float* __restrict__ kh,
    const float* __restrict__ vh, const int* __restrict__ typed_ids,
    const float* __restrict__ bias, float* __restrict__ attn) {
  __shared__ __align__(16) float Klds[16 * KPAD];     // K row-major [k][d]
  __shared__ __align__(16) float VldsT[64 * VTPAD];   // V transposed [n][k]
  __shared__ __align__(16) float Plds[4 * 16 * PPAD]; // per-wave P tiles

  const int qtile = blockIdx.x;   // 0..31
  const int bh    = blockIdx.y;   // 0..15
  const int bb    = bh >> 3;
  const int h     = bh & 7;

  const int tid = threadIdx.x, wave = tid >> 5, lane = tid & 31;
  const int lh = lane & 15, lg = lane >> 4;
  const int wq = qtile * 64 + wave * 16;        // first q row of this wave

  // Preload A fragments of the 3 typed queries (16 q-rows x 64 dh each).
  v2f aq[3][16];
#pragma unroll
  for (int t = 0; t < 3; ++t) {
    const float* qb = qt + (((size_t)t * 16 + bh) * L_ + wq + lh) * DH_;
#pragma unroll
    for (int kb = 0; kb < 16; ++kb)
      aq[t][kb] = *(const v2f*)(qb + kb * 4 + 2 * lg);
  }

  v8f acc_o[4] = {};
  float m_run[8], l_run[8];
#pragma unroll
  for (int v = 0; v < 8; ++v) { m_run[v] = -1e30f; l_run[v] = 0.0f; }

  const float* kbase = kh + (size_t)bh * L_ * DH_;
  const float* vbase = vh + (size_t)bh * L_ * DH_;
  float* Pw = Plds + wave * 16 * PPAD;

  for (int k0 = 0; k0 < L_; k0 += 16) {
    // cooperative stage: K (16x64 row-major), V (transposed to [n][k])
    {
      const int c4 = (tid & 15) * 4;
      const int r0 = tid >> 4;                  // 0..7
#pragma unroll
      for (int it = 0; it < 2; ++it) {
        const int r = r0 + it * 8;
        const float4 kv = *(const float4*)(kbase + (size_t)(k0 + r) * DH_ + c4);
        *(float4*)(Klds + r * KPAD + c4) = kv;
        const float4 vv = *(const float4*)(vbase + (size_t)(k0 + r) * DH_ + c4);
        VldsT[(c4 + 0) * VTPAD + r] = vv.x;
        VldsT[(c4 + 1) * VTPAD + r] = vv.y;
        VldsT[(c4 + 2) * VTPAD + r] = vv.z;
        VldsT[(c4 + 3) * VTPAD + r] = vv.w;
      }
    }
    __syncthreads();

    // scores for all 3 types: S_t = qt_t (16x64) @ K^T (64x16)
    v8f s0 = {}, s1 = {}, s2 = {};
#pragma unroll
    for (int kb = 0; kb < 16; ++kb) {
      // B fragment: b[i] = K[k0+n][d+i], n = lh (contiguous b64)
      const v2f bk_ = *(const v2f*)(Klds + lh * KPAD + kb * 4 + 2 * lg);
      s0 = wmma_f32(aq[0][kb], bk_, s0);
      s1 = wmma_f32(aq[1][kb], bk_, s1);
      s2 = wmma_f32(aq[2][kb], bk_, s2);
    }

    // typed select + bias  (element (m,n): lane = n + 16*(m/8), vgpr = m%8)
    const float bval = bias[(size_t)bb * L_ + k0 + lh];
    float sc[8];
#pragma unroll
    for (int v = 0; v < 8; ++v) {
      const int gq = wq + v + 8 * lg;
      const int tt = typed_ids[((size_t)bb * L_ + gq) * L_ + k0 + lh];
      const float sv = (tt == 0) ? s0[v] : (tt == 1) ? s1[v] : s2[v];
      sc[v] = sv + bval;
    }

    // row max across the 16 lanes holding this row
    float tmax[8];
#pragma unroll
    for (int v = 0; v < 8; ++v) tmax[v] = sc[v];
#pragma unroll
    for (int mask = 8; mask >= 1; mask >>= 1)
#pragma unroll
      for (int v = 0; v < 8; ++v)
        tmax[v] = fmaxf(tmax[v], __shfl_xor(tmax[v], mask, 32));

    float corr[8], rsum[8];
#pragma unroll
    for (int v = 0; v < 8; ++v) {
      const float nm = fmaxf(m_run[v], tmax[v]);
      corr[v] = __expf(m_run[v] - nm);
      sc[v]   = __expf(sc[v] - nm);   // now holds P
      m_run[v] = nm;
      rsum[v] = sc[v];
    }
#pragma unroll
    for (int mask = 8; mask >= 1; mask >>= 1)
#pragma unroll
      for (int v = 0; v < 8; ++v)
        rsum[v] += __shfl_xor(rsum[v], mask, 32);
#pragma unroll
    for (int v = 0; v < 8; ++v)
      l_run[v] = l_run[v] * corr[v] + rsum[v];

    // rescale O accumulators
#pragma unroll
    for (int j = 0; j < 4; ++j)
#pragma unroll
      for (int v = 0; v < 8; ++v)
        acc_o[j][v] = acc_o[j][v] * corr[v];

    // C-layout -> A-layout for P via per-wave LDS region (same-wave DS ops
    // are in-order; no barrier needed)
#pragma unroll
    for (int v = 0; v < 8; ++v)
      Pw[(v + 8 * lg) * PPAD + lh] = sc[v];

    // O += P (16x16) @ V (16x64);  V B-fragments are contiguous b64 reads
#pragma unroll
    for (int kb = 0; kb < 4; ++kb) {
      const int kd = kb * 4 + 2 * lg;
      const v2f ap = *(const v2f*)(Pw + lh * PPAD + kd);
#pragma unroll
      for (int j = 0; j < 4; ++j) {
        const v2f bv_ = *(const v2f*)(VldsT + (j * 16 + lh) * VTPAD + kd);
        acc_o[j] = wmma_f32(ap, bv_, acc_o[j]);
      }
    }
    __syncthreads();   // protect Klds/VldsT before next tile load
  }

  // normalize and write combined-head layout [B, L, HID]
#pragma unroll
  for (int v = 0; v < 8; ++v) {
    const float inv = 1.0f / l_run[v];
    const int gq = wq + v + 8 * lg;
#pragma unroll
    for (int j = 0; j < 4; ++j)
      attn[((size_t)bb * L_ + gq) * HID_ + h * 64 + j * 16 + lh] =
          acc_o[j][v] * inv;
  }
}

// ---------------------------------------------------------------------------
// Kernel 4: output projection  out = attn @ Wo + bo
// ---------------------------------------------------------------------------
__global__ __launch_bounds__(128) void oproj_kernel(
    const float* __restrict__ X, const float* __restrict__ Wo,
    const float* __restrict__ bo, float* __restrict__ out) {
  __shared__ __align__(16) float As[128 * APAD];
  __shared__ __align__(16) float BsT[64 * BTPAD];

  const int ntile = blockIdx.x;  // 0..7
  const int mtile = blockIdx.y;  // 0..31

  v8f acc[2][4] = {};
  gemm_tile_128x64(X + (size_t)mtile * 128 * HID_, HID_,
                   Wo + ntile * 64, HID_, HID_, As, BsT, acc);

  const int tid = threadIdx.x, wave = tid >> 5, lane = tid & 31;
  const int lh = lane & 15, lg = lane >> 4;
#pragma unroll
  for (int s = 0; s < 2; ++s) {
#pragma unroll
    for (int j = 0; j < 4; ++j) {
      const int gc = ntile * 64 + j * 16 + lh;
      const float bv_ = bo[gc];
#pragma unroll
      for (int v = 0; v < 8; ++v) {
        const int gr = mtile * 128 + s * 64 + wave * 16 + v + 8 * lg;
        out[(size_t)gr * HID_ + gc] = acc[s][j][v] + bv_;
      }
    }
  }
}

// ---------------------------------------------------------------------------
extern "C" void kernel_launch(void* const* d_in, const int* in_sizes, int n_in,
                              void* d_out, int out_size, void* d_ws,
                              size_t ws_size, hipStream_t stream) {
  (void)in_sizes; (void)n_in; (void)out_size; (void)ws_size;

  const float* query = (const float*)d_in[0];
  const float* bias  = (const float*)d_in[1];
  const int*   tids  = (const int*)d_in[2];
  const float* Wq = (const float*)d_in[3];
  const float* bq = (const float*)d_in[4];
  const float* Wk = (const float*)d_in[5];
  const float* bk = (const float*)d_in[6];
  const float* Wv = (const float*)d_in[7];
  const float* bv = (const float*)d_in[8];
  const float* Wo = (const float*)d_in[9];
  const float* bo = (const float*)d_in[10];
  const float* tw = (const float*)d_in[11];
  float* out = (float*)d_out;

  float* ws = (float*)d_ws;
  const size_t HEAD = (size_t)B_ * NH_ * L_ * DH_;   // 2,097,152 floats
  float* qtbuf = ws;                 // 3*HEAD  typed queries
  float* khbuf = ws + 3 * HEAD;      // 1*HEAD
  float* vhbuf = ws + 4 * HEAD;      // 1*HEAD
  float* qhbuf = ws + 5 * HEAD;      // 1*HEAD  (reused as attn output)
  float* attnb = qhbuf;              // qh dead after qt_kernel -> reuse

  proj_kernel<<<dim3(8, 32, 3), 128, 0, stream>>>(
      query, Wq, bq, Wk, bk, Wv, bv, qhbuf, khbuf, vhbuf);
  qt_kernel<<<dim3(16, 48), 128, 0, stream>>>(qhbuf, tw, qtbuf);
  attn_kernel<<<dim3(32, 16), 128, 0, stream>>>(
      qtbuf, khbuf, vhbuf, tids, bias, attnb);
  oproj_kernel<<<dim3(8, 32), 128, 0, stream>>>(attnb, Wo, bo, out);
}